// SpGAT_e2t_37641093382708
// MI455X (gfx1250) — compile-verified
//
#include <hip/hip_runtime.h>
#include <math.h>

// ---------------------------------------------------------------------------
// SpGAT e2t (KBGAT-style) for MI455X / gfx1250 (wave32, WMMA).
//
// Strategy (roofline):
//   * All dense projections as fp32 WMMA GEMMs (V_WMMA_F32_16X16X4_F32):
//       P1 = X1 @ a[:, :d1]^T   (node-level)
//       P2 = X2 @ a[:, d1:d1+d2]^T
//       P3 = E  @ a[:, d1+d2:]^T  (edge-level, layer 1)
//     Layer 2: e2 = out_rel[edge_type] has only 200 distinct rows, so its
//     projection is a tiny 200xK GEMM + per-edge gather.
//   * Edge phase: one wave32 per edge; m = P1[src]+P2[dst]+P3[*];
//     score = exp(-leaky_relu(m.v)); atomic scatter-add num/den.
//   * Normalize + ELU elementwise.
//
// v2 changes:
//   * Branch-free A-fragment loads (row index clamped once; OOB rows read row
//     0 and are dropped by the store guard) -> no per-iteration EXEC dance,
//     loads form clauses and pipeline across the unrolled K loop.
//   * 32x64 C strip per wave (2 A fragments x 4 N-tiles, 8 accumulators):
//     every B fragment feeds two WMMAs, halving B-side VMEM per output.
// ---------------------------------------------------------------------------

typedef __attribute__((ext_vector_type(2))) float v2f;
typedef __attribute__((ext_vector_type(8))) float v8f;

#define ALPHA 0.2f

// C[M x (gridDim.y*64)] = A[M x K] * B_rows^T, where
//   C[m][n] = sum_k A[m*lda + k] * B[n*ldb + boff + k]
// One wave computes a 32x64 strip of C via 8 WMMA accumulators.
template <int K>
__global__ __launch_bounds__(256) void gemm_abt_wmma(
    const float* __restrict__ A, int lda,
    const float* __restrict__ B, int ldb, int boff,
    float* __restrict__ C, int ldc, int M)
{
    const int wave = threadIdx.x >> 5;
    const int lane = threadIdx.x & 31;
    const int half = lane >> 4;   // 0: K pair {0,1}; 1: K pair {2,3}
    const int l    = lane & 15;

    const int m0 = (blockIdx.x * 8 + wave) * 32;   // 32 rows per wave
    const int n0 = blockIdx.y * 64;

    // Branch-free OOB handling: clamp row to 0 (contributes only to C rows
    // that the store guard drops), keep EXEC uniform at every load & WMMA.
    int arow0 = m0 + l;      if (arow0 >= M) arow0 = 0;
    int arow1 = m0 + 16 + l; if (arow1 >= M) arow1 = 0;
    const float* ap0 = A + (size_t)arow0 * lda + 2 * half;
    const float* ap1 = A + (size_t)arow1 * lda + 2 * half;
    const float* bp0 = B + (size_t)(n0 +  0 + l) * ldb + boff + 2 * half;
    const float* bp1 = B + (size_t)(n0 + 16 + l) * ldb + boff + 2 * half;
    const float* bp2 = B + (size_t)(n0 + 32 + l) * ldb + boff + 2 * half;
    const float* bp3 = B + (size_t)(n0 + 48 + l) * ldb + boff + 2 * half;

    v8f acc00 = {}, acc01 = {}, acc02 = {}, acc03 = {};
    v8f acc10 = {}, acc11 = {}, acc12 = {}, acc13 = {};

#pragma unroll
    for (int k0 = 0; k0 < K; k0 += 4) {
        v2f a0; a0.x = ap0[k0]; a0.y = ap0[k0 + 1];
        v2f a1; a1.x = ap1[k0]; a1.y = ap1[k0 + 1];
        v2f b0; b0.x = bp0[k0]; b0.y = bp0[k0 + 1];
        v2f b1; b1.x = bp1[k0]; b1.y = bp1[k0 + 1];
        v2f b2; b2.x = bp2[k0]; b2.y = bp2[k0 + 1];
        v2f b3; b3.x = bp3[k0]; b3.y = bp3[k0 + 1];

        acc00 = __builtin_amdgcn_wmma_f32_16x16x4_f32(false, a0, false, b0, (short)0, acc00, false, false);
        acc01 = __builtin_amdgcn_wmma_f32_16x16x4_f32(false, a0, false, b1, (short)0, acc01, false, false);
        acc02 = __builtin_amdgcn_wmma_f32_16x16x4_f32(false, a0, false, b2, (short)0, acc02, false, false);
        acc03 = __builtin_amdgcn_wmma_f32_16x16x4_f32(false, a0, false, b3, (short)0, acc03, false, false);
        acc10 = __builtin_amdgcn_wmma_f32_16x16x4_f32(false, a1, false, b0, (short)0, acc10, false, false);
        acc11 = __builtin_amdgcn_wmma_f32_16x16x4_f32(false, a1, false, b1, (short)0, acc11, false, false);
        acc12 = __builtin_amdgcn_wmma_f32_16x16x4_f32(false, a1, false, b2, (short)0, acc12, false, false);
        acc13 = __builtin_amdgcn_wmma_f32_16x16x4_f32(false, a1, false, b3, (short)0, acc13, false, false);
    }

    // C/D layout: VGPR r -> row base + r + 8*half, col n0 + l (per 16-col tile)
#pragma unroll
    for (int r = 0; r < 8; ++r) {
        int row = m0 + r + 8 * half;
        if (row < M) {
            float* cp = C + (size_t)row * ldc + n0 + l;
            cp[0]  = acc00[r];
            cp[16] = acc01[r];
            cp[32] = acc02[r];
            cp[48] = acc03[r];
        }
    }
#pragma unroll
    for (int r = 0; r < 8; ++r) {
        int row = m0 + 16 + r + 8 * half;
        if (row < M) {
            float* cp = C + (size_t)row * ldc + n0 + l;
            cp[0]  = acc10[r];
            cp[16] = acc11[r];
            cp[32] = acc12[r];
            cp[48] = acc13[r];
        }
    }
}

// Layer-1 edge phase, D=64. One wave32 per edge, 2 components per lane.
__global__ __launch_bounds__(256) void edge_scatter_d64(
    const float* __restrict__ P1, const float* __restrict__ P2,
    const float* __restrict__ P3,
    const int* __restrict__ src, const int* __restrict__ dst,
    const float* __restrict__ v, int useSrcSeg,
    float* __restrict__ num, float* __restrict__ den, int E)
{
    int gid  = blockIdx.x * blockDim.x + threadIdx.x;
    int e    = gid >> 5;
    int lane = threadIdx.x & 31;
    if (e >= E) return;

    int s = src[e], d = dst[e];
    int seg = useSrcSeg ? s : d;

    float m0 = P1[(size_t)s * 64 + lane]      + P2[(size_t)d * 64 + lane]      + P3[(size_t)e * 64 + lane];
    float m1 = P1[(size_t)s * 64 + lane + 32] + P2[(size_t)d * 64 + lane + 32] + P3[(size_t)e * 64 + lane + 32];

    float part = m0 * v[lane] + m1 * v[lane + 32];
#pragma unroll
    for (int off = 16; off > 0; off >>= 1)
        part += __shfl_xor(part, off, 32);

    float lr = part > 0.0f ? part : ALPHA * part;
    float sc = expf(-lr);

    atomicAdd(&num[(size_t)seg * 64 + lane],      m0 * sc);
    atomicAdd(&num[(size_t)seg * 64 + lane + 32], m1 * sc);
    if (lane == 0) atomicAdd(&den[seg], sc);
}

// Layer-2 edge phase, D=128; edge feature indexed by edge_type (200 rows).
__global__ __launch_bounds__(256) void edge_scatter_d128(
    const float* __restrict__ P1, const float* __restrict__ P2,
    const float* __restrict__ P3rel,
    const int* __restrict__ src, const int* __restrict__ dst,
    const int* __restrict__ etype,
    const float* __restrict__ v, int useSrcSeg,
    float* __restrict__ num, float* __restrict__ den, int E)
{
    int gid  = blockIdx.x * blockDim.x + threadIdx.x;
    int e    = gid >> 5;
    int lane = threadIdx.x & 31;
    if (e >= E) return;

    int s = src[e], d = dst[e], t = etype[e];
    int seg = useSrcSeg ? s : d;

    float m[4];
    float part = 0.0f;
#pragma unroll
    for (int j = 0; j < 4; ++j) {
        int c = lane + 32 * j;
        m[j] = P1[(size_t)s * 128 + c] + P2[(size_t)d * 128 + c] + P3rel[(size_t)t * 128 + c];
        part += m[j] * v[c];
    }
#pragma unroll
    for (int off = 16; off > 0; off >>= 1)
        part += __shfl_xor(part, off, 32);

    float lr = part > 0.0f ? part : ALPHA * part;
    float sc = expf(-lr);

#pragma unroll
    for (int j = 0; j < 4; ++j)
        atomicAdd(&num[(size_t)seg * 128 + lane + 32 * j], m[j] * sc);
    if (lane == 0) atomicAdd(&den[seg], sc);
}

// h = elu(num/den'), scattered into the concat buffer (layer-1, concat=True).
__global__ void normalize_elu_concat64(
    const float* __restrict__ num, const float* __restrict__ den,
    float* __restrict__ xbuf, int nseg, int ldx, int colOff)
{
    int i = blockIdx.x * blockDim.x + threadIdx.x;
    if (i >= nseg * 64) return;
    int n = i >> 6, c = i & 63;
    float dn = den[n];
    dn = (dn == 0.0f) ? 1.0f : dn;
    float h = num[i] / dn;
    h = h > 0.0f ? h : expm1f(h);
    xbuf[(size_t)n * ldx + colOff + c] = h;
}

// Final: out = elu(num/den'), D=128 (reference applies elu after layer 2).
__global__ void normalize_elu_out128(
    const float* __restrict__ num, const float* __restrict__ den,
    float* __restrict__ out, int nseg)
{
    int i = blockIdx.x * blockDim.x + threadIdx.x;
    if (i >= nseg * 128) return;
    int n = i >> 7;
    float dn = den[n];
    dn = (dn == 0.0f) ? 1.0f : dn;
    float h = num[i] / dn;
    out[i] = h > 0.0f ? h : expm1f(h);
}

// out_rel[r][j] = sum_k rel[r][k] * W[k][j]   (200x128x64, negligible cost)
__global__ void rel_w_kernel(const float* __restrict__ rel,
                             const float* __restrict__ W,
                             float* __restrict__ out_rel, int nrel)
{
    int idx = blockIdx.x * blockDim.x + threadIdx.x;
    if (idx >= nrel * 128) return;
    int r = idx >> 7, j = idx & 127;
    float acc = 0.0f;
#pragma unroll
    for (int k = 0; k < 64; ++k)
        acc += rel[r * 64 + k] * W[k * 128 + j];
    out_rel[idx] = acc;
}

static inline unsigned cdiv(long long a, long long b) { return (unsigned)((a + b - 1) / b); }

extern "C" void kernel_launch(void* const* d_in, const int* in_sizes, int n_in,
                              void* d_out, int out_size, void* d_ws, size_t ws_size,
                              hipStream_t stream)
{
    const int D = 64, HO = 128;
    const float* ent  = (const float*)d_in[0];
    const float* typ  = (const float*)d_in[1];
    const float* rel  = (const float*)d_in[2];
    const int*   el   = (const int*)  d_in[3];
    const int*   etyp = (const int*)  d_in[4];
    const float* eemb = (const float*)d_in[5];
    const float* a_e0 = (const float*)d_in[6];
    const float* v_e0 = (const float*)d_in[7];
    const float* a_t0 = (const float*)d_in[8];
    const float* v_t0 = (const float*)d_in[9];
    const float* a_e1 = (const float*)d_in[10];
    const float* v_e1 = (const float*)d_in[11];
    const float* a_t1 = (const float*)d_in[12];
    const float* v_t1 = (const float*)d_in[13];
    const float* a_eo = (const float*)d_in[14];
    const float* v_eo = (const float*)d_in[15];
    const float* a_to = (const float*)d_in[16];
    const float* v_to = (const float*)d_in[17];
    const float* W    = (const float*)d_in[18];

    const int N_ENT = in_sizes[0] / D;
    const int N_TYP = in_sizes[1] / D;
    const int N_REL = in_sizes[2] / D;
    const int E     = in_sizes[4];
    const int* src  = el;
    const int* dst  = el + E;

    // Workspace layout (floats)
    float* x1   = (float*)d_ws;                 // N_ENT*128
    float* x2   = x1  + (size_t)N_ENT * 128;    // N_TYP*128
    float* P1   = x2  + (size_t)N_TYP * 128;    // N_ENT*128 (sized for layer 2)
    float* P2   = P1  + (size_t)N_ENT * 128;    // N_TYP*128
    float* P3   = P2  + (size_t)N_TYP * 128;    // E*64 (layer-1 edge projection)
    float* P3o  = P3  + (size_t)E * 64;         // N_REL*128
    float* numb = P3o + (size_t)N_REL * 128;    // N_ENT*128
    float* den  = numb+ (size_t)N_ENT * 128;    // N_ENT

    float* o1   = (float*)d_out;
    float* o2   = o1 + (size_t)N_ENT * HO;
    float* orel = o2 + (size_t)N_TYP * HO;

    const dim3 blk(256);
    const unsigned edgeBlocks = cdiv((long long)E * 32, 256);

    // ---------------- Layer 1: 4 streams (head x {entity,type} aggregation) --
    struct Stream1 { const float* a; const float* v; int useSrc; float* xbuf; int nseg; int colOff; };
    Stream1 s1[4] = {
        { a_e0, v_e0, 1, x1, N_ENT, 0  },
        { a_t0, v_t0, 0, x2, N_TYP, 0  },
        { a_e1, v_e1, 1, x1, N_ENT, 64 },
        { a_t1, v_t1, 0, x2, N_TYP, 64 },
    };

    for (int i = 0; i < 4; ++i) {
        const Stream1& st = s1[i];
        // Projections (WMMA GEMMs): Din = 192, slices at boff 0 / 64 / 128
        gemm_abt_wmma<64><<<dim3(cdiv(N_ENT, 256), 1), blk, 0, stream>>>(
            ent, D, st.a, 192, 0, P1, 64, N_ENT);
        gemm_abt_wmma<64><<<dim3(cdiv(N_TYP, 256), 1), blk, 0, stream>>>(
            typ, D, st.a, 192, 64, P2, 64, N_TYP);
        gemm_abt_wmma<64><<<dim3(cdiv(E, 256), 1), blk, 0, stream>>>(
            eemb, D, st.a, 192, 128, P3, 64, E);

        hipMemsetAsync(numb, 0, (size_t)st.nseg * 64 * sizeof(float), stream);
        hipMemsetAsync(den,  0, (size_t)st.nseg * sizeof(float), stream);

        edge_scatter_d64<<<edgeBlocks, blk, 0, stream>>>(
            P1, P2, P3, src, dst, st.v, st.useSrc, numb, den, E);

        normalize_elu_concat64<<<cdiv((long long)st.nseg * 64, 256), blk, 0, stream>>>(
            numb, den, st.xbuf, st.nseg, 128, st.colOff);
    }

    // ---------------- out_rel = relation_embed @ W (written straight to d_out)
    rel_w_kernel<<<cdiv((long long)N_REL * 128, 256), blk, 0, stream>>>(rel, W, orel, N_REL);

    // ---------------- Layer 2: 2 streams (entity-agg -> o1, type-agg -> o2) --
    struct Stream2 { const float* a; const float* v; int useSrc; float* out; int nseg; };
    Stream2 s2[2] = {
        { a_eo, v_eo, 1, o1, N_ENT },
        { a_to, v_to, 0, o2, N_TYP },
    };

    for (int i = 0; i < 2; ++i) {
        const Stream2& st = s2[i];
        // Din = 384, slices at boff 0 / 128 / 256; K = 128, N = 128 (grid.y = 2)
        gemm_abt_wmma<128><<<dim3(cdiv(N_ENT, 256), 2), blk, 0, stream>>>(
            x1, 128, st.a, 384, 0, P1, 128, N_ENT);
        gemm_abt_wmma<128><<<dim3(cdiv(N_TYP, 256), 2), blk, 0, stream>>>(
            x2, 128, st.a, 384, 128, P2, 128, N_TYP);
        gemm_abt_wmma<128><<<dim3(cdiv(N_REL, 256), 2), blk, 0, stream>>>(
            orel, 128, st.a, 384, 256, P3o, 128, N_REL);

        hipMemsetAsync(numb, 0, (size_t)st.nseg * 128 * sizeof(float), stream);
        hipMemsetAsync(den,  0, (size_t)st.nseg * sizeof(float), stream);

        edge_scatter_d128<<<edgeBlocks, blk, 0, stream>>>(
            P1, P2, P3o, src, dst, etyp, st.v, st.useSrc, numb, den, E);

        normalize_elu_out128<<<cdiv((long long)st.nseg * 128, 256), blk, 0, stream>>>(
            numb, den, st.out, st.nseg);
    }
}